// ReconstructionLoss_31001073943294
// MI455X (gfx1250) — compile-verified
//
#include <hip/hip_runtime.h>

// ---------------------------------------------------------------------------
// Problem constants (from reference): B=64, C=256, feats 64x64 spatial,
// pixel tensors 64x3x256x256.
// ---------------------------------------------------------------------------
#define NPIX      12582912            // 64*3*256*256
#define NPIX4     (NPIX / 4)          // float4 count = 3145728
#define REC_BLOCKS 1024
#define BATCH     64
#define CHANS     256
#define HW        4096
#define NBC       (BATCH * CHANS)     // 16384

// Workspace layout (float offsets)
#define OFF_EMB   0                   // [64*256]  GAP embedding
#define OFF_DOT   16384               // [16384]   per-(b,c) dot(fa,fb)
#define OFF_NA    32768               // [16384]   per-(b,c) sum fa^2
#define OFF_NB    49152               // [16384]   per-(b,c) sum fb^2
#define OFF_REC   65536               // [1024]    per-block rec partials
#define OFF_CON   66560               // [1024]    per-block con partials
#define OFF_COS   67584               // [64]      per-batch cosine

typedef float v2f __attribute__((ext_vector_type(2)));
typedef float v4f __attribute__((ext_vector_type(4)));   // native vector: OK for nontemporal builtins
typedef float v8f __attribute__((ext_vector_type(8)));

// ---------------------------------------------------------------------------
// Kernel 1: streaming pass over out_a/out_b/target -> per-block partial sums
// of (out_b - target)^2 and (out_a - out_b)^2. Deterministic tree reduction.
// ---------------------------------------------------------------------------
__global__ __launch_bounds__(256)
void rec_con_partial(const v4f* __restrict__ a4, const v4f* __restrict__ b4,
                     const v4f* __restrict__ t4,
                     float* __restrict__ rec_part, float* __restrict__ con_part) {
  __shared__ float sr[256];
  __shared__ float sc[256];
  const int tid = threadIdx.x;
  float rec = 0.f, con = 0.f;
  for (int i = blockIdx.x * 256 + tid; i < NPIX4; i += REC_BLOCKS * 256) {
    v4f va = __builtin_nontemporal_load(&a4[i]);
    v4f vb = __builtin_nontemporal_load(&b4[i]);
    v4f vt = __builtin_nontemporal_load(&t4[i]);
    float dx = vb.x - vt.x, dy = vb.y - vt.y, dz = vb.z - vt.z, dw = vb.w - vt.w;
    rec += dx * dx + dy * dy + dz * dz + dw * dw;
    dx = va.x - vb.x; dy = va.y - vb.y; dz = va.z - vb.z; dw = va.w - vb.w;
    con += dx * dx + dy * dy + dz * dz + dw * dw;
  }
  sr[tid] = rec; sc[tid] = con;
  __syncthreads();
  for (int s = 128; s > 0; s >>= 1) {
    if (tid < s) { sr[tid] += sr[tid + s]; sc[tid] += sc[tid + s]; }
    __syncthreads();
  }
  if (tid == 0) { rec_part[blockIdx.x] = sr[0]; con_part[blockIdx.x] = sc[0]; }
}

// ---------------------------------------------------------------------------
// Kernel 2: one block per (b,c) channel (4096 contiguous floats each).
// Single streaming pass over feat_a/feat_b producing:
//   dot(fa,fb), sum fa^2, sum fb^2 (partials per (b,c)),
//   GAP embedding emb[b,c] = mean(fb over HW).
// ---------------------------------------------------------------------------
__global__ __launch_bounds__(256)
void feat_pass(const v4f* __restrict__ fa4, const v4f* __restrict__ fb4,
               float* __restrict__ ws) {
  const int bc = blockIdx.x;                     // 0..16383
  const int tid = threadIdx.x;
  const v4f* pa = fa4 + (size_t)bc * (HW / 4);
  const v4f* pb = fb4 + (size_t)bc * (HW / 4);
  float sab = 0.f, saa = 0.f, sbb = 0.f, sb = 0.f;
#pragma unroll
  for (int it = 0; it < 4; ++it) {
    v4f a = __builtin_nontemporal_load(&pa[tid + it * 256]);
    v4f b = __builtin_nontemporal_load(&pb[tid + it * 256]);
    sab += a.x * b.x + a.y * b.y + a.z * b.z + a.w * b.w;
    saa += a.x * a.x + a.y * a.y + a.z * a.z + a.w * a.w;
    sbb += b.x * b.x + b.y * b.y + b.z * b.z + b.w * b.w;
    sb  += b.x + b.y + b.z + b.w;
  }
  __shared__ float s0[256], s1[256], s2[256], s3[256];
  s0[tid] = sab; s1[tid] = saa; s2[tid] = sbb; s3[tid] = sb;
  __syncthreads();
  for (int s = 128; s > 0; s >>= 1) {
    if (tid < s) {
      s0[tid] += s0[tid + s]; s1[tid] += s1[tid + s];
      s2[tid] += s2[tid + s]; s3[tid] += s3[tid + s];
    }
    __syncthreads();
  }
  if (tid == 0) {
    ws[OFF_DOT + bc] = s0[0];
    ws[OFF_NA  + bc] = s1[0];
    ws[OFF_NB  + bc] = s2[0];
    ws[OFF_EMB + bc] = s3[0] * (1.0f / (float)HW);   // GAP
  }
}

// ---------------------------------------------------------------------------
// Kernel 3: per-batch cosine from the 256 channel partials.
// ---------------------------------------------------------------------------
__global__ __launch_bounds__(256)
void cos_per_batch(float* __restrict__ ws) {
  const int b = blockIdx.x;
  const int t = threadIdx.x;
  __shared__ float sd[256], sa[256], sb[256];
  sd[t] = ws[OFF_DOT + b * CHANS + t];
  sa[t] = ws[OFF_NA  + b * CHANS + t];
  sb[t] = ws[OFF_NB  + b * CHANS + t];
  __syncthreads();
  for (int s = 128; s > 0; s >>= 1) {
    if (t < s) { sd[t] += sd[t + s]; sa[t] += sa[t + s]; sb[t] += sb[t + s]; }
    __syncthreads();
  }
  if (t == 0) {
    float na = sqrtf(sa[0]); if (na < 1e-8f) na = 1e-8f;
    float nb = sqrtf(sb[0]); if (nb < 1e-8f) nb = 1e-8f;
    ws[OFF_COS + b] = sd[0] / (na * nb);
  }
}

// ---------------------------------------------------------------------------
// Kernel 4: finalize. 512 threads = 16 waves.
//  (a) Gram = E * E^T (64x64, K=256) via V_WMMA_F32_16X16X4_F32,
//      one 16x16 tile per wave. A (16x4) and B (4x16 = E^T slice) share the
//      same lane striping when sourced from row-major E:
//        lane L holds row base+(L&15); VGPR0/1 = K+0/K+1 (lanes 0-15),
//        K+2/K+3 (lanes 16-31).
//  (b) Reduce rec/con partials + cosine sum.
//  (c) Batch-hard triplet mining from Gram (64 rows, one per thread).
// ---------------------------------------------------------------------------
__global__ __launch_bounds__(512)
void finalize(const float* __restrict__ ws, const int* __restrict__ labels,
              float* __restrict__ out) {
  __shared__ float gram[BATCH * BATCH];   // 16 KB
  __shared__ float red[512];
  __shared__ float red2[512];
  const int tid  = threadIdx.x;
  const int wave = tid >> 5;
  const int lane = tid & 31;

  // --- (a) WMMA Gram matrix ---
  const float* emb = ws + OFF_EMB;
  const int i0 = (wave >> 2) * 16;              // tile row base
  const int j0 = (wave & 3) * 16;               // tile col base
  const int koff = (lane < 16) ? 0 : 2;         // A/B K sub-offset per half-wave
  const int ra = (i0 + (lane & 15)) * CHANS + koff;
  const int rb = (j0 + (lane & 15)) * CHANS + koff;
  v8f acc = {};
  for (int k = 0; k < CHANS; k += 4) {
    v2f a, b;
    a.x = emb[ra + k]; a.y = emb[ra + k + 1];
    b.x = emb[rb + k]; b.y = emb[rb + k + 1];
    acc = __builtin_amdgcn_wmma_f32_16x16x4_f32(false, a, false, b,
                                                (short)0, acc, false, false);
  }
  // D layout: VGPR r -> M = r (lanes 0-15) / M = 8+r (lanes 16-31), N = lane&15
  {
    const int mbase = i0 + ((lane < 16) ? 0 : 8);
    const int col   = j0 + (lane & 15);
#pragma unroll
    for (int r = 0; r < 8; ++r) gram[(mbase + r) * BATCH + col] = acc[r];
  }

  // --- (b) reduce rec/con partials (1024 each) ---
  red[tid]  = ws[OFF_REC + tid] + ws[OFF_REC + tid + 512];
  red2[tid] = ws[OFF_CON + tid] + ws[OFF_CON + tid + 512];
  __syncthreads();
  for (int s = 256; s > 0; s >>= 1) {
    if (tid < s) { red[tid] += red[tid + s]; red2[tid] += red2[tid + s]; }
    __syncthreads();
  }
  const float loss_rec = red[0] * (1.0f / (float)NPIX);
  const float loss_con = red2[0] * (1.0f / (float)NPIX);
  __syncthreads();

  // cosine mean
  red[tid] = (tid < BATCH) ? ws[OFF_COS + tid] : 0.f;
  __syncthreads();
  for (int s = 256; s > 0; s >>= 1) {
    if (tid < s) red[tid] += red[tid + s];
    __syncthreads();
  }
  const float loss_align = 1.0f - red[0] * (1.0f / (float)BATCH);
  __syncthreads();

  // --- (c) batch-hard triplet from Gram (already barriered above) ---
  float per_row = 0.f, validf = 0.f;
  if (tid < BATCH) {
    const int li = labels[tid];
    const float sqi = gram[tid * BATCH + tid];
    float hp = -3.402823466e38f, hn = 3.402823466e38f;
    bool has_p = false, has_n = false;
    for (int j = 0; j < BATCH; ++j) {
      float d2 = sqi + gram[j * BATCH + j] - 2.0f * gram[tid * BATCH + j];
      float dist = sqrtf(fmaxf(d2, 0.0f) + 1e-12f);
      bool same = (labels[j] == li);
      if (same && j != tid) { hp = fmaxf(hp, dist); has_p = true; }
      if (!same)            { hn = fminf(hn, dist); has_n = true; }
    }
    if (has_p && has_n) { per_row = fmaxf(hp - hn + 0.5f, 0.f); validf = 1.f; }
  }
  red[tid] = per_row; red2[tid] = validf;
  __syncthreads();
  for (int s = 256; s > 0; s >>= 1) {
    if (tid < s) { red[tid] += red[tid + s]; red2[tid] += red2[tid + s]; }
    __syncthreads();
  }
  if (tid == 0) {
    const float loss_sep = red[0] / (red2[0] + 1e-8f);
    out[0] = 1.0f * loss_rec + 1.0f * loss_con
           + 0.5f * loss_align + 2.0f * loss_sep;
  }
}

// ---------------------------------------------------------------------------
extern "C" void kernel_launch(void* const* d_in, const int* in_sizes, int n_in,
                              void* d_out, int out_size, void* d_ws, size_t ws_size,
                              hipStream_t stream) {
  (void)in_sizes; (void)n_in; (void)out_size; (void)ws_size;
  const v4f* out_a  = (const v4f*)d_in[0];
  const v4f* out_b  = (const v4f*)d_in[1];
  const v4f* target = (const v4f*)d_in[2];
  const v4f* feat_a = (const v4f*)d_in[3];
  const v4f* feat_b = (const v4f*)d_in[4];
  const int* labels = (const int*)d_in[5];
  float* ws  = (float*)d_ws;
  float* out = (float*)d_out;

  rec_con_partial<<<REC_BLOCKS, 256, 0, stream>>>(out_a, out_b, target,
                                                  ws + OFF_REC, ws + OFF_CON);
  feat_pass<<<NBC, 256, 0, stream>>>(feat_a, feat_b, ws);
  cos_per_batch<<<BATCH, 256, 0, stream>>>(ws);
  finalize<<<1, 512, 0, stream>>>(ws, labels, out);
}